// GraphCPI_GATGCN_36850819400360
// MI455X (gfx1250) — compile-verified
//
#include <hip/hip_runtime.h>
#include <hip/hip_bf16.h>
#include <math.h>

// ---------------- problem constants ----------------
#define N_NODES 16384
#define N_EDGES 65536
#define ET      (N_EDGES + N_NODES)   // edges + self loops = 81920
#define N_GRAPHS 512
#define F_XD    78
#define HEADS   10
#define HID     (F_XD * HEADS)        // 780
#define SEQ     1000
#define EMB     100
#define NF      32
#define KS      8
#define CLEN    (EMB - KS + 1)        // 93
#define OUT_DIM 128
#define DCAT    (2 * HID)             // 1560

typedef float v2f __attribute__((ext_vector_type(2)));
typedef float v8f __attribute__((ext_vector_type(8)));

// ---------------- helpers ----------------
__device__ inline int esrc(const int* ei, int i) { return i < N_EDGES ? ei[i] : i - N_EDGES; }
__device__ inline int edst(const int* ei, int i) { return i < N_EDGES ? ei[N_EDGES + i] : i - N_EDGES; }

__device__ inline void atomicMaxF(float* addr, float val) {
    int* ai = (int*)addr;
    int old = __float_as_int(*addr);
    while (__int_as_float(old) < val) {
        int assumed = old;
        old = atomicCAS(ai, assumed, __float_as_int(val));
        if (old == assumed) break;
    }
}

__device__ inline v8f wmma_f32(v2f a, v2f b, v8f c) {
    return __builtin_amdgcn_wmma_f32_16x16x4_f32(false, a, false, b, (short)0, c, false, false);
}

// ---------------- generic WMMA f32 GEMM ----------------
// C[M,N](ldc) = op(A[M,K](lda) @ B[K,N](ldb) + bias)
// One wave computes a 32x32 macro-tile (2x2 of 16x16 V_WMMA_F32_16X16X4_F32).
// Interior tiles run a branch-free loop; edge tiles use a guarded 16x16 path.

__device__ void gemm_tile_guarded(const float* __restrict__ A, int lda,
                                  const float* __restrict__ B, int ldb,
                                  const float* __restrict__ bias,
                                  float* __restrict__ C, int ldc,
                                  int M, int N, int K, int relu,
                                  int m0, int n0, int half, int l16)
{
    v8f acc = {};
    for (int k = 0; k < K; k += 4) {
        v2f a, b;
        const int ar  = m0 + l16;
        const int ak0 = k + half * 2;
        a[0] = (ar < M && ak0     < K) ? A[(size_t)ar * lda + ak0]     : 0.0f;
        a[1] = (ar < M && ak0 + 1 < K) ? A[(size_t)ar * lda + ak0 + 1] : 0.0f;
        const int bc = n0 + l16;
        b[0] = (bc < N && ak0     < K) ? B[(size_t)(ak0)     * ldb + bc] : 0.0f;
        b[1] = (bc < N && ak0 + 1 < K) ? B[(size_t)(ak0 + 1) * ldb + bc] : 0.0f;
        acc = wmma_f32(a, b, acc);
    }
    const int col = n0 + l16;
    if (col < N) {
        const float bv = bias ? bias[col] : 0.0f;
        #pragma unroll
        for (int r = 0; r < 8; ++r) {
            const int row = m0 + r + half * 8;
            if (row < M) {
                float v = acc[r] + bv;
                if (relu) v = fmaxf(v, 0.0f);
                C[(size_t)row * ldc + col] = v;
            }
        }
    }
}

__global__ void gemm_f32_wmma(const float* __restrict__ A, int lda,
                              const float* __restrict__ B, int ldb,
                              const float* __restrict__ bias,
                              float* __restrict__ C, int ldc,
                              int M, int N, int K, int relu)
{
    const int n0   = blockIdx.x * 32;
    const int m0   = blockIdx.y * 32;
    const int lane = threadIdx.x;
    const int half = lane >> 4;
    const int l16  = lane & 15;

    if (m0 + 32 <= M && n0 + 32 <= N) {
        // ---- interior fast path: no predication around WMMA ----
        v8f acc00 = {}, acc01 = {}, acc10 = {}, acc11 = {};
        const float* Arow0 = A + (size_t)(m0 + l16) * lda;
        const float* Arow1 = A + (size_t)(m0 + 16 + l16) * lda;
        const float* Bc0   = B + n0 + l16;
        const float* Bc1   = B + n0 + 16 + l16;
        const int K4 = K & ~3;
        for (int k = 0; k < K4; k += 4) {
            const int ak = k + half * 2;
            v2f a0, a1, b0, b1;
            a0[0] = Arow0[ak];  a0[1] = Arow0[ak + 1];
            a1[0] = Arow1[ak];  a1[1] = Arow1[ak + 1];
            b0[0] = Bc0[(size_t)ak * ldb];  b0[1] = Bc0[(size_t)(ak + 1) * ldb];
            b1[0] = Bc1[(size_t)ak * ldb];  b1[1] = Bc1[(size_t)(ak + 1) * ldb];
            __builtin_prefetch(&Bc0[(size_t)(ak + 8) * ldb], 0, 3); // global_prefetch_b8
            acc00 = wmma_f32(a0, b0, acc00);
            acc01 = wmma_f32(a0, b1, acc01);
            acc10 = wmma_f32(a1, b0, acc10);
            acc11 = wmma_f32(a1, b1, acc11);
        }
        if (K4 < K) {   // K-remainder (e.g. K=78): guarded single step
            const int ak = K4 + half * 2;
            v2f a0, a1, b0, b1;
            a0[0] = (ak     < K) ? Arow0[ak]     : 0.0f;
            a0[1] = (ak + 1 < K) ? Arow0[ak + 1] : 0.0f;
            a1[0] = (ak     < K) ? Arow1[ak]     : 0.0f;
            a1[1] = (ak + 1 < K) ? Arow1[ak + 1] : 0.0f;
            b0[0] = (ak     < K) ? Bc0[(size_t)ak * ldb]       : 0.0f;
            b0[1] = (ak + 1 < K) ? Bc0[(size_t)(ak + 1) * ldb] : 0.0f;
            b1[0] = (ak     < K) ? Bc1[(size_t)ak * ldb]       : 0.0f;
            b1[1] = (ak + 1 < K) ? Bc1[(size_t)(ak + 1) * ldb] : 0.0f;
            acc00 = wmma_f32(a0, b0, acc00);
            acc01 = wmma_f32(a0, b1, acc01);
            acc10 = wmma_f32(a1, b0, acc10);
            acc11 = wmma_f32(a1, b1, acc11);
        }
        // epilogue (fully in-bounds)
        const int c0 = n0 + l16, c1 = n0 + 16 + l16;
        const float bv0 = bias ? bias[c0] : 0.0f;
        const float bv1 = bias ? bias[c1] : 0.0f;
        #pragma unroll
        for (int r = 0; r < 8; ++r) {
            const int row0 = m0 + r + half * 8;
            const int row1 = row0 + 16;
            float v00 = acc00[r] + bv0, v01 = acc01[r] + bv1;
            float v10 = acc10[r] + bv0, v11 = acc11[r] + bv1;
            if (relu) {
                v00 = fmaxf(v00, 0.0f); v01 = fmaxf(v01, 0.0f);
                v10 = fmaxf(v10, 0.0f); v11 = fmaxf(v11, 0.0f);
            }
            C[(size_t)row0 * ldc + c0] = v00;
            C[(size_t)row0 * ldc + c1] = v01;
            C[(size_t)row1 * ldc + c0] = v10;
            C[(size_t)row1 * ldc + c1] = v11;
        }
    } else {
        // ---- edge path: 4 guarded 16x16 tiles ----
        #pragma unroll
        for (int tm = 0; tm < 2; ++tm)
            #pragma unroll
            for (int tn = 0; tn < 2; ++tn)
                if (m0 + tm * 16 < M && n0 + tn * 16 < N)
                    gemm_tile_guarded(A, lda, B, ldb, bias, C, ldc, M, N, K, relu,
                                      m0 + tm * 16, n0 + tn * 16, half, l16);
    }
}

// ---------------- elementwise / fill ----------------
__global__ void fill_kernel(float* p, float v, int n) {
    int i = blockIdx.x * blockDim.x + threadIdx.x;
    if (i < n) p[i] = v;
}

__global__ void bias_relu_kernel(float* x, const float* bias, int rows, int cols) {
    int i = blockIdx.x * blockDim.x + threadIdx.x;
    if (i < rows * cols) x[i] = fmaxf(x[i] + bias[i % cols], 0.0f);
}

// ---------------- GAT attention ----------------
__global__ void att_kernel(const float* __restrict__ xh,
                           const float* __restrict__ att_src,
                           const float* __restrict__ att_dst,
                           float* __restrict__ a_src, float* __restrict__ a_dst)
{
    int idx = blockIdx.x * blockDim.x + threadIdx.x;   // n*HEADS + h
    if (idx >= N_NODES * HEADS) return;
    int n = idx / HEADS, h = idx % HEADS;
    const float* xrow = xh + (size_t)n * HID + h * F_XD;
    float s = 0.0f, d = 0.0f;
    for (int c = 0; c < F_XD; ++c) {
        float v = xrow[c];
        s += v * att_src[h * F_XD + c];
        d += v * att_dst[h * F_XD + c];
    }
    a_src[idx] = s;
    a_dst[idx] = d;
}

__global__ void edge_e_kernel(const int* __restrict__ ei,
                              const float* __restrict__ a_src,
                              const float* __restrict__ a_dst,
                              float* __restrict__ e)
{
    int i = blockIdx.x * blockDim.x + threadIdx.x;
    if (i >= ET) return;
    int s = esrc(ei, i), d = edst(ei, i);
    #pragma unroll
    for (int h = 0; h < HEADS; ++h) {
        float v = a_src[s * HEADS + h] + a_dst[d * HEADS + h];
        e[i * HEADS + h] = v > 0.0f ? v : 0.2f * v;   // leaky_relu 0.2
    }
}

__global__ void seg_max_kernel(const int* __restrict__ ei,
                               const float* __restrict__ e,
                               float* __restrict__ emax)
{
    int i = blockIdx.x * blockDim.x + threadIdx.x;
    if (i >= ET) return;
    int d = edst(ei, i);
    #pragma unroll
    for (int h = 0; h < HEADS; ++h)
        atomicMaxF(&emax[d * HEADS + h], e[i * HEADS + h]);
}

__global__ void edge_exp_kernel(const int* __restrict__ ei,
                                float* __restrict__ e,            // in: e, out: exp(e - emax)
                                const float* __restrict__ emax,
                                float* __restrict__ den)
{
    int i = blockIdx.x * blockDim.x + threadIdx.x;
    if (i >= ET) return;
    int d = edst(ei, i);
    #pragma unroll
    for (int h = 0; h < HEADS; ++h) {
        float v = __expf(e[i * HEADS + h] - emax[d * HEADS + h]);
        e[i * HEADS + h] = v;
        atomicAdd(&den[d * HEADS + h], v);
    }
}

// one 256-thread block per edge: h[dst, f] += xh[src, f] * coef(edge, f/78)
__global__ void gat_aggregate_kernel(const int* __restrict__ ei,
                                     const float* __restrict__ xh,
                                     const float* __restrict__ ee,
                                     const float* __restrict__ den,
                                     float* __restrict__ h)
{
    int i = blockIdx.x;
    int s = esrc(ei, i), d = edst(ei, i);
    for (int f = threadIdx.x; f < HID; f += blockDim.x) {
        int hh = f / F_XD;
        float coef = ee[i * HEADS + hh] / den[d * HEADS + hh];
        atomicAdd(&h[(size_t)d * HID + f], xh[(size_t)s * HID + f] * coef);
    }
}

// ---------------- GCN ----------------
__global__ void deg_kernel(const int* __restrict__ ei, float* __restrict__ deg) {
    int i = blockIdx.x * blockDim.x + threadIdx.x;
    if (i >= ET) return;
    atomicAdd(&deg[edst(ei, i)], 1.0f);
}

__global__ void dinv_kernel(float* deg) {
    int i = blockIdx.x * blockDim.x + threadIdx.x;
    if (i < N_NODES) deg[i] = rsqrtf(fmaxf(deg[i], 1.0f));
}

__global__ void gcn_aggregate_kernel(const int* __restrict__ ei,
                                     const float* __restrict__ hw,
                                     const float* __restrict__ dinv,
                                     float* __restrict__ g)
{
    int i = blockIdx.x;
    int s = esrc(ei, i), d = edst(ei, i);
    float w = dinv[s] * dinv[d];
    for (int f = threadIdx.x; f < HID; f += blockDim.x)
        atomicAdd(&g[(size_t)d * HID + f], hw[(size_t)s * HID + f] * w);
}

// ---------------- pooling ----------------
__global__ void pool_kernel(const int* __restrict__ batch,
                            const float* __restrict__ g,
                            float* __restrict__ dcat,   // gmp lives in cols [0,780)
                            float* __restrict__ gsum)
{
    int n = blockIdx.x;
    int b = batch[n];
    for (int f = threadIdx.x; f < HID; f += blockDim.x) {
        float v = g[(size_t)n * HID + f];
        atomicMaxF(&dcat[(size_t)b * DCAT + f], v);
        atomicAdd(&gsum[(size_t)b * HID + f], v);
    }
}

__global__ void cnt_kernel(const int* __restrict__ batch, float* __restrict__ cnt) {
    int n = blockIdx.x * blockDim.x + threadIdx.x;
    if (n < N_NODES) atomicAdd(&cnt[batch[n]], 1.0f);
}

__global__ void gap_kernel(const float* __restrict__ gsum,
                           const float* __restrict__ cnt,
                           float* __restrict__ dcat)
{
    int i = blockIdx.x * blockDim.x + threadIdx.x;
    if (i >= N_GRAPHS * HID) return;
    int b = i / HID, f = i % HID;
    dcat[(size_t)b * DCAT + HID + f] = gsum[i] / fmaxf(cnt[b], 1.0f);
}

// ---------------- protein conv branch: WMMA implicit GEMM ----------------
// c[b,f,t] = conv_b[f] + sum_{ci,k} emb_w[target[b,ci], t+k] * conv_w[f,ci,k]
// Per protein b:  C[32 x 93] = A[32 x 8000] @ X[8000 x 93]
//   A[f][ci*8+k] = conv_w[f*8000 + ci*8 + k]  (contiguous in K -> direct global loads)
//   X[(ci*8+k)][t] = emb_w[target[b,ci]*100 + t + k]   (built in LDS per 40-ci chunk)
// Block: 192 threads = 6 waves; each wave owns 2 adjacent 16-wide column tiles of C
// (tile ids 2w, 2w+1 share the A fragment). Accumulators persist across 25 chunks.
#define CI_CHUNK 40
#define KC       (CI_CHUNK * KS)      // 320 = per-chunk K
#define NPAD     96                   // 93 padded to 6*16
__global__ void conv_wmma_kernel(const int* __restrict__ target,
                                 const float* __restrict__ emb_w,
                                 const float* __restrict__ conv_w,
                                 const float* __restrict__ conv_b,
                                 float* __restrict__ crs)
{
    __shared__ float rows[CI_CHUNK * EMB];   // 16 KB gathered embedding rows
    __shared__ float xcol[KC * NPAD];        // 120 KB im2col panel
    const int b    = blockIdx.x;
    const int tid  = threadIdx.x;
    const int wave = tid >> 5;               // 0..5
    const int lane = tid & 31;
    const int half = lane >> 4;
    const int l16  = lane & 15;

    const int ti0 = 2 * wave;                // tiles 2w, 2w+1 (same tm row)
    const int tm  = ti0 / 6;                 // 0 or 1  -> filter rows f = tm*16..
    const int tn0 = ti0 % 6;                 // column tile pair tn0, tn0+1
    const float* Arow = conv_w + (size_t)(tm * 16 + l16) * (SEQ * KS);

    v8f acc0 = {}, acc1 = {};
    for (int ci0 = 0; ci0 < SEQ; ci0 += CI_CHUNK) {
        __syncthreads();   // previous chunk's WMMA reads of xcol are done
        for (int e2 = tid; e2 < CI_CHUNK * EMB; e2 += blockDim.x) {
            int r = e2 / EMB, c2 = e2 % EMB;
            rows[e2] = emb_w[(size_t)target[b * SEQ + ci0 + r] * EMB + c2];
        }
        __syncthreads();
        for (int e2 = tid; e2 < KC * NPAD; e2 += blockDim.x) {
            int r = e2 / NPAD, t = e2 % NPAD;        // r = (ci_local*8 + k)
            int ci = r >> 3, k = r & 7;
            xcol[e2] = (t < CLEN) ? rows[ci * EMB + t + k] : 0.0f;
        }
        __syncthreads();
        const float* Ap = Arow + ci0 * KS;           // A[f][ci0*8 + ...]
        for (int kk = 0; kk < KC; kk += 4) {
            const int ak = kk + half * 2;
            v2f a, b0, b1;
            a[0]  = Ap[ak];  a[1] = Ap[ak + 1];
            b0[0] = xcol[(ak)     * NPAD + tn0 * 16 + l16];
            b0[1] = xcol[(ak + 1) * NPAD + tn0 * 16 + l16];
            b1[0] = xcol[(ak)     * NPAD + tn0 * 16 + 16 + l16];
            b1[1] = xcol[(ak + 1) * NPAD + tn0 * 16 + 16 + l16];
            acc0 = wmma_f32(a, b0, acc0);
            acc1 = wmma_f32(a, b1, acc1);
        }
    }
    // epilogue: crs[b, f*93 + t], guard t < 93
    const int t0 = tn0 * 16 + l16;
    const int t1 = t0 + 16;
    #pragma unroll
    for (int r = 0; r < 8; ++r) {
        const int f = tm * 16 + r + half * 8;
        const float bv = conv_b[f];
        if (t0 < CLEN) crs[(size_t)b * (NF * CLEN) + f * CLEN + t0] = acc0[r] + bv;
        if (t1 < CLEN) crs[(size_t)b * (NF * CLEN) + f * CLEN + t1] = acc1[r] + bv;
    }
}

// ---------------- host side ----------------
static inline float* ws_alloc(char*& p, size_t nfloats) {
    float* r = (float*)p;
    p += ((nfloats * sizeof(float) + 255) / 256) * 256;
    return r;
}

static inline void launch_gemm(const float* A, int lda, const float* B, int ldb,
                               const float* bias, float* C, int ldc,
                               int M, int N, int K, int relu, hipStream_t stream) {
    dim3 grid((N + 31) / 32, (M + 31) / 32);
    gemm_f32_wmma<<<grid, 32, 0, stream>>>(A, lda, B, ldb, bias, C, ldc, M, N, K, relu);
}

extern "C" void kernel_launch(void* const* d_in, const int* in_sizes, int n_in,
                              void* d_out, int out_size, void* d_ws, size_t ws_size,
                              hipStream_t stream) {
    const float* x          = (const float*)d_in[0];
    const int*   edge_index = (const int*)  d_in[1];
    const int*   batch      = (const int*)  d_in[2];
    const int*   target     = (const int*)  d_in[3];
    const float* gat_w      = (const float*)d_in[4];
    const float* gat_att_src= (const float*)d_in[5];
    const float* gat_att_dst= (const float*)d_in[6];
    const float* gat_b      = (const float*)d_in[7];
    const float* gcn_w      = (const float*)d_in[8];
    const float* gcn_b      = (const float*)d_in[9];
    const float* fcg1_w     = (const float*)d_in[10];
    const float* fcg1_b     = (const float*)d_in[11];
    const float* fcg2_w     = (const float*)d_in[12];
    const float* fcg2_b     = (const float*)d_in[13];
    const float* emb_w      = (const float*)d_in[14];
    const float* conv_w     = (const float*)d_in[15];
    const float* conv_b     = (const float*)d_in[16];
    const float* fcxt_w     = (const float*)d_in[17];
    const float* fcxt_b     = (const float*)d_in[18];
    const float* mlp1_w     = (const float*)d_in[19];
    const float* mlp1_b     = (const float*)d_in[20];
    const float* mlp2_w     = (const float*)d_in[21];
    const float* mlp2_b     = (const float*)d_in[22];
    const float* mlp3_w     = (const float*)d_in[23];
    const float* mlp3_b     = (const float*)d_in[24];
    float* out = (float*)d_out;

    // workspace carve-out (xh buffer is reused for g after GAT aggregation)
    char* p = (char*)d_ws;
    float* xh_g  = ws_alloc(p, (size_t)N_NODES * HID);   // xh, later g
    float* h     = ws_alloc(p, (size_t)N_NODES * HID);
    float* hw    = ws_alloc(p, (size_t)N_NODES * HID);
    float* a_src = ws_alloc(p, (size_t)N_NODES * HEADS);
    float* a_dst = ws_alloc(p, (size_t)N_NODES * HEADS);
    float* e     = ws_alloc(p, (size_t)ET * HEADS);      // e, later exp(e - emax)
    float* emax  = ws_alloc(p, (size_t)N_NODES * HEADS);
    float* den   = ws_alloc(p, (size_t)N_NODES * HEADS);
    float* deg   = ws_alloc(p, (size_t)N_NODES);         // later dinv (in place)
    float* dcat  = ws_alloc(p, (size_t)N_GRAPHS * DCAT);
    float* gsum  = ws_alloc(p, (size_t)N_GRAPHS * HID);
    float* cnt   = ws_alloc(p, (size_t)N_GRAPHS);
    float* d1    = ws_alloc(p, (size_t)N_GRAPHS * 1500);
    float* ocat  = ws_alloc(p, (size_t)N_GRAPHS * 2 * OUT_DIM);
    float* crs   = ws_alloc(p, (size_t)N_GRAPHS * NF * CLEN);
    float* o1    = ws_alloc(p, (size_t)N_GRAPHS * 1024);
    float* o2    = ws_alloc(p, (size_t)N_GRAPHS * 512);

    const int TB = 256;
    const int edge_blocks = (ET + TB - 1) / TB;

    // ---- GAT ----
    launch_gemm(x, F_XD, gat_w, HID, nullptr, xh_g, HID, N_NODES, HID, F_XD, 0, stream);
    att_kernel<<<(N_NODES * HEADS + TB - 1) / TB, TB, 0, stream>>>(xh_g, gat_att_src, gat_att_dst, a_src, a_dst);
    edge_e_kernel<<<edge_blocks, TB, 0, stream>>>(edge_index, a_src, a_dst, e);
    fill_kernel<<<(N_NODES * HEADS + TB - 1) / TB, TB, 0, stream>>>(emax, -INFINITY, N_NODES * HEADS);
    seg_max_kernel<<<edge_blocks, TB, 0, stream>>>(edge_index, e, emax);
    hipMemsetAsync(den, 0, (size_t)N_NODES * HEADS * sizeof(float), stream);
    edge_exp_kernel<<<edge_blocks, TB, 0, stream>>>(edge_index, e, emax, den);
    hipMemsetAsync(h, 0, (size_t)N_NODES * HID * sizeof(float), stream);
    gat_aggregate_kernel<<<ET, TB, 0, stream>>>(edge_index, xh_g, e, den, h);
    bias_relu_kernel<<<((size_t)N_NODES * HID + TB - 1) / TB, TB, 0, stream>>>(h, gat_b, N_NODES, HID);

    // ---- GCN ----
    launch_gemm(h, HID, gcn_w, HID, nullptr, hw, HID, N_NODES, HID, HID, 0, stream);
    hipMemsetAsync(deg, 0, (size_t)N_NODES * sizeof(float), stream);
    deg_kernel<<<edge_blocks, TB, 0, stream>>>(edge_index, deg);
    dinv_kernel<<<(N_NODES + TB - 1) / TB, TB, 0, stream>>>(deg);
    hipMemsetAsync(xh_g, 0, (size_t)N_NODES * HID * sizeof(float), stream);   // g
    gcn_aggregate_kernel<<<ET, TB, 0, stream>>>(edge_index, hw, deg, xh_g);
    bias_relu_kernel<<<((size_t)N_NODES * HID + TB - 1) / TB, TB, 0, stream>>>(xh_g, gcn_b, N_NODES, HID);

    // ---- pooling ----
    fill_kernel<<<(N_GRAPHS * DCAT + TB - 1) / TB, TB, 0, stream>>>(dcat, -INFINITY, N_GRAPHS * DCAT);
    hipMemsetAsync(gsum, 0, (size_t)N_GRAPHS * HID * sizeof(float), stream);
    hipMemsetAsync(cnt, 0, (size_t)N_GRAPHS * sizeof(float), stream);
    pool_kernel<<<N_NODES, TB, 0, stream>>>(batch, xh_g, dcat, gsum);
    cnt_kernel<<<(N_NODES + TB - 1) / TB, TB, 0, stream>>>(batch, cnt);
    gap_kernel<<<(N_GRAPHS * HID + TB - 1) / TB, TB, 0, stream>>>(gsum, cnt, dcat);

    // ---- graph FC branch (d -> ocat[:, :128]) ----
    launch_gemm(dcat, DCAT, fcg1_w, 1500, fcg1_b, d1, 1500, N_GRAPHS, 1500, DCAT, 1, stream);
    launch_gemm(d1, 1500, fcg2_w, OUT_DIM, fcg2_b, ocat, 2 * OUT_DIM, N_GRAPHS, OUT_DIM, 1500, 0, stream);

    // ---- protein branch (p -> ocat[:, 128:256]) ----
    conv_wmma_kernel<<<N_GRAPHS, 192, 0, stream>>>(target, emb_w, conv_w, conv_b, crs);
    launch_gemm(crs, NF * CLEN, fcxt_w, OUT_DIM, fcxt_b, ocat + OUT_DIM, 2 * OUT_DIM,
                N_GRAPHS, OUT_DIM, NF * CLEN, 0, stream);

    // ---- MLP head ----
    launch_gemm(ocat, 2 * OUT_DIM, mlp1_w, 1024, mlp1_b, o1, 1024, N_GRAPHS, 1024, 2 * OUT_DIM, 1, stream);
    launch_gemm(o1, 1024, mlp2_w, 512, mlp2_b, o2, 512, N_GRAPHS, 512, 1024, 1, stream);
    launch_gemm(o2, 512, mlp3_w, 1, mlp3_b, out, 1, N_GRAPHS, 1, 512, 0, stream);
}